// Baseline_Loss_12781822673485
// MI455X (gfx1250) — compile-verified
//
#include <hip/hip_runtime.h>
#include <math.h>

// Problem shape (fixed by the reference): B=1, S=5, Q=512, T=64, D=2048
#define QN   512
#define SN   5
#define SPAD 16      // pad support classes to one WMMA tile width
#define TN   64
#define DN   2048
#define EPSF 1e-8f

typedef __attribute__((ext_vector_type(2))) float v2f;
typedef __attribute__((ext_vector_type(8))) float v8f;

// ---------------------------------------------------------------------------
// Kernel 1: pool supp over T -> sp_pad [16 x 2048] (rows >=5 zeroed), and
// sn[n] = ||sp[n]||.  Tiny (2.6 MB read), one block per padded class.
// ---------------------------------------------------------------------------
__global__ void pool_supp_kernel(const float* __restrict__ supp,
                                 float* __restrict__ sp_pad,
                                 float* __restrict__ sn) {
    const int n   = blockIdx.x;     // 0..15
    const int tid = threadIdx.x;    // 0..511, one float4 column each
    float4 acc = make_float4(0.f, 0.f, 0.f, 0.f);
    if (n < SN) {
        const float4* src = reinterpret_cast<const float4*>(supp + (size_t)n * TN * DN);
        #pragma unroll 4
        for (int t = 0; t < TN; ++t) {
            float4 v = src[t * (DN / 4) + tid];
            acc.x += v.x; acc.y += v.y; acc.z += v.z; acc.w += v.w;
        }
        const float s = 1.0f / (float)TN;
        acc.x *= s; acc.y *= s; acc.z *= s; acc.w *= s;
    }
    reinterpret_cast<float4*>(sp_pad + (size_t)n * DN)[tid] = acc;

    __shared__ float red[512];
    red[tid] = acc.x * acc.x + acc.y * acc.y + acc.z * acc.z + acc.w * acc.w;
    __syncthreads();
    for (int s2 = 256; s2 > 0; s2 >>= 1) {
        if (tid < s2) red[tid] += red[tid + s2];
        __syncthreads();
    }
    if (tid == 0) sn[n] = sqrtf(red[0]);
}

// ---------------------------------------------------------------------------
// Kernel 2: pool query over T -> qp [512 x 2048], qn[q] = ||qp[q]||.
// This is the bandwidth-bound kernel: 268 MB streamed, fully coalesced
// float4 loads (512 threads == 2048 floats per row), ~11.5 us at 23.3 TB/s.
// ---------------------------------------------------------------------------
__global__ void pool_query_kernel(const float* __restrict__ query,
                                  float* __restrict__ qp,
                                  float* __restrict__ qn) {
    const int q   = blockIdx.x;     // 0..511
    const int tid = threadIdx.x;    // 0..511
    const float4* src = reinterpret_cast<const float4*>(query + (size_t)q * TN * DN);
    float4 acc = make_float4(0.f, 0.f, 0.f, 0.f);
    #pragma unroll 4
    for (int t = 0; t < TN; ++t) {
        float4 v = src[t * (DN / 4) + tid];
        acc.x += v.x; acc.y += v.y; acc.z += v.z; acc.w += v.w;
    }
    const float s = 1.0f / (float)TN;
    acc.x *= s; acc.y *= s; acc.z *= s; acc.w *= s;
    reinterpret_cast<float4*>(qp + (size_t)q * DN)[tid] = acc;

    __shared__ float red[512];
    red[tid] = acc.x * acc.x + acc.y * acc.y + acc.z * acc.z + acc.w * acc.w;
    __syncthreads();
    for (int s2 = 256; s2 > 0; s2 >>= 1) {
        if (tid < s2) red[tid] += red[tid + s2];
        __syncthreads();
    }
    if (tid == 0) qn[q] = sqrtf(red[0]);
}

// ---------------------------------------------------------------------------
// Kernel 3: num = qp (512x2048) x sp_pad^T (2048x16) via V_WMMA_F32_16X16X4_F32
// (exact fp32 -> matches the fp32 reference einsum). One 16x16 output tile
// per block; K=2048 split across 8 waves (64 WMMAs each), partial C tiles
// reduced through LDS. Then dist = num / max(qn*sn, eps); writes 1-dist to
// d_out and dist to workspace for the CE kernel.
//
// f32 WMMA operand layout (ISA 7.12.2):
//   A 16x4: lane m = lane%16 selects row M; VGPR j holds K = j + 2*(lane/16)
//   B 4x16: lane n = lane%16 selects col N; VGPR j holds K = j + 2*(lane/16)
//   C/D 16x16: VGPR v, half h = lane/16 -> (M = v + 8h, N = lane%16)
// ---------------------------------------------------------------------------
__global__ void wmma_cosine_kernel(const float* __restrict__ qp,
                                   const float* __restrict__ sp_pad,
                                   const float* __restrict__ qn,
                                   const float* __restrict__ sn,
                                   float* __restrict__ dist_ws,
                                   float* __restrict__ out) {
    const int q0   = blockIdx.x * 16;
    const int tid  = threadIdx.x;   // 0..255 (8 waves)
    const int w    = tid >> 5;      // wave id 0..7 -> K slice
    const int lane = tid & 31;
    const int m    = lane & 15;     // A row / B col within tile
    const int h    = lane >> 4;     // lane half

    const float* arow = qp     + (size_t)(q0 + m) * DN;
    const float* brow = sp_pad + (size_t)m * DN;

    v8f acc = {};
    const int kbase = w * (DN / 8);        // 256 K-values per wave
    #pragma unroll 8
    for (int i = 0; i < (DN / 8) / 4; ++i) {   // 64 steps of K=4
        const int k = kbase + i * 4 + 2 * h;
        v2f a = *reinterpret_cast<const v2f*>(arow + k);
        v2f b = *reinterpret_cast<const v2f*>(brow + k);
        acc = __builtin_amdgcn_wmma_f32_16x16x4_f32(
            /*neg_a=*/false, a, /*neg_b=*/false, b,
            /*c_mod=*/(short)0, acc, /*reuse_a=*/false, /*reuse_b=*/false);
    }

    __shared__ float part[8][256];
    #pragma unroll
    for (int v = 0; v < 8; ++v) {
        const int M = v + 8 * h;
        part[w][M * 16 + m] = acc[v];
    }
    __syncthreads();

    // 256 threads cover the 256 tile elements; sum the 8 K-partials.
    float sum = 0.f;
    #pragma unroll
    for (int ww = 0; ww < 8; ++ww) sum += part[ww][tid];

    const int M = tid >> 4;
    const int N = tid & 15;
    const int q = q0 + M;
    if (N < SN) {
        const float d = sum / fmaxf(qn[q] * sn[N], EPSF);
        dist_ws[q * SN + N]  = d;
        out[1 + q * SN + N]  = 1.0f - d;   // second tuple element, row-major [Q,S]
    }
}

// ---------------------------------------------------------------------------
// Kernel 4: loss = mean_q ( logsumexp(dist[q,:]) - dist[q, y_q] ).
// One block of 512 threads, one row each, tree reduction.
// ---------------------------------------------------------------------------
__global__ void ce_loss_kernel(const float* __restrict__ dist_ws,
                               const int* __restrict__ ys,
                               float* __restrict__ out) {
    const int q = threadIdx.x;     // 0..511
    const float* row = dist_ws + q * SN;
    float d0 = row[0], d1 = row[1], d2 = row[2], d3 = row[3], d4 = row[4];
    float mx  = fmaxf(fmaxf(fmaxf(d0, d1), fmaxf(d2, d3)), d4);
    float lse = logf(expf(d0 - mx) + expf(d1 - mx) + expf(d2 - mx)
                   + expf(d3 - mx) + expf(d4 - mx)) + mx;
    int y = ys[q];
    y = y < 0 ? 0 : (y > SN - 1 ? SN - 1 : y);
    float contrib = lse - row[y];

    __shared__ float red[512];
    red[q] = contrib;
    __syncthreads();
    for (int s = 256; s > 0; s >>= 1) {
        if (q < s) red[q] += red[q + s];
        __syncthreads();
    }
    if (q == 0) out[0] = red[0] / (float)QN;
}

// ---------------------------------------------------------------------------
extern "C" void kernel_launch(void* const* d_in, const int* in_sizes, int n_in,
                              void* d_out, int out_size, void* d_ws, size_t ws_size,
                              hipStream_t stream) {
    (void)in_sizes; (void)n_in; (void)out_size; (void)ws_size;
    const float* supp  = (const float*)d_in[0];   // [1,5,64,2048] f32
    const float* query = (const float*)d_in[1];   // [1,512,64,2048] f32
    const int*   ys    = (const int*)d_in[2];     // [512] int

    float* out = (float*)d_out;                   // [1 + 512*5] f32
    float* ws  = (float*)d_ws;

    // Workspace layout (floats):
    float* sp_pad = ws;                           // 16*2048     = 32768
    float* sn     = sp_pad + SPAD * DN;           // 16
    float* qn     = sn + SPAD;                    // 512
    float* dist   = qn + QN;                      // 512*5 (alloc 512*8 for alignment)
    float* qp     = dist + QN * 8;                // 512*2048  (~4.3 MB total)

    pool_supp_kernel <<<SPAD,   512, 0, stream>>>(supp, sp_pad, sn);
    pool_query_kernel<<<QN,     512, 0, stream>>>(query, qp, qn);
    wmma_cosine_kernel<<<QN/16, 256, 0, stream>>>(qp, sp_pad, qn, sn, dist, out);
    ce_loss_kernel   <<<1,      QN,  0, stream>>>(dist, ys, out);
}